// Selective_Scan_52080773431762
// MI455X (gfx1250) — compile-verified
//
#include <hip/hip_runtime.h>
#include <math.h>
#include <stdint.h>

typedef __attribute__((ext_vector_type(2))) float v2f;
typedef __attribute__((ext_vector_type(8))) float v8f;

#define BATCHN   2
#define SEQ      4096
#define DINNER   2048
#define DSTATE   16
#define DTRANK   64
#define C96      (DTRANK + 2 * DSTATE)   // 96
#define ROWS     (BATCHN * SEQ)          // 8192
#define NCHUNK   32
#define CHUNK    (SEQ / NCHUNK)          // 128

// ---------------------------------------------------------------------------
// K1: x_dbl[row][c] = sum_d x[row][d] * Wxp[c][d]   (M=8192, N=96, K=2048)
//     rows of prompt added into c = 80..95 (the C-matrix of the SSM).
// f32 WMMA 16x16x4.  A reg r <-> K=r+2g, B reg r <-> K=r+2g, C reg r <-> M=r+8g
// ---------------------------------------------------------------------------
__global__ __launch_bounds__(256) void k_xproj(const float* __restrict__ x,
                                               const float* __restrict__ Wxp,
                                               const float* __restrict__ prompt,
                                               float* __restrict__ xdbl) {
    const int lane = threadIdx.x & 31;
    const int wave = threadIdx.x >> 5;
    const int m = lane & 15;          // M-row / N-col within tile
    const int g = lane >> 4;          // half-wave select
    const int mtile = blockIdx.x * 8 + wave;
    const int row0 = mtile * 16;

    v8f acc[6] = {};
    const float* xrow = x + (size_t)(row0 + m) * DINNER + 2 * g;

#pragma unroll 1
    for (int kk = 0; kk < DINNER; kk += 4) {
        v2f a = *(const v2f*)(xrow + kk);
#pragma unroll
        for (int j = 0; j < 6; ++j) {
            v2f bm = *(const v2f*)(Wxp + (size_t)(j * 16 + m) * DINNER + kk + 2 * g);
            acc[j] = __builtin_amdgcn_wmma_f32_16x16x4_f32(
                false, a, false, bm, (short)0, acc[j], false, false);
        }
    }

#pragma unroll
    for (int j = 0; j < 6; ++j) {
#pragma unroll
        for (int r = 0; r < 8; ++r) {
            const int rowr = row0 + r + 8 * g;
            const int c = j * 16 + m;
            float v = acc[j][r];
            if (j == 5) v += prompt[(size_t)rowr * DSTATE + m];   // Cs += prompt
            xdbl[(size_t)rowr * C96 + c] = v;
        }
    }
}

// ---------------------------------------------------------------------------
// K2: delta[row][d] = softplus( sum_r xdbl[row][r] * Wdt[d][r] + bias[d] )
//     (M=8192, N=2048, K=64).  One wave: 1 m-tile x 8 n-tiles.
// ---------------------------------------------------------------------------
__global__ __launch_bounds__(256) void k_delta(const float* __restrict__ xdbl,
                                               const float* __restrict__ Wdt,
                                               const float* __restrict__ bias,
                                               float* __restrict__ delta) {
    const int lane = threadIdx.x & 31;
    const int wave = threadIdx.x >> 5;
    const int m = lane & 15;
    const int g = lane >> 4;
    const int w = blockIdx.x * 8 + wave;
    const int mtile = w >> 4;          // 512 m-tiles
    const int ngrp  = w & 15;          // 16 groups of 8 n-tiles
    const int row0 = mtile * 16;
    const int d0 = ngrp * 128;

    v8f acc[8] = {};
    const float* arow = xdbl + (size_t)(row0 + m) * C96 + 2 * g;

#pragma unroll 1
    for (int kk = 0; kk < DTRANK; kk += 4) {
        v2f a = *(const v2f*)(arow + kk);
#pragma unroll
        for (int j = 0; j < 8; ++j) {
            v2f bm = *(const v2f*)(Wdt + (size_t)(d0 + j * 16 + m) * DTRANK + kk + 2 * g);
            acc[j] = __builtin_amdgcn_wmma_f32_16x16x4_f32(
                false, a, false, bm, (short)0, acc[j], false, false);
        }
    }

#pragma unroll
    for (int j = 0; j < 8; ++j) {
        const int d = d0 + j * 16 + m;
        const float bs = bias[d];
#pragma unroll
        for (int r = 0; r < 8; ++r) {
            const int rowr = row0 + r + 8 * g;
            float v = acc[j][r] + bs;
            float sp = (v > 20.0f) ? v : log1pf(__expf(v));
            delta[(size_t)rowr * DINNER + d] = sp;
        }
    }
}

// ---------------------------------------------------------------------------
// Async-stage the chunk's B/C slab (CHUNK x 32 floats = 16 KB) into LDS with
// gfx1250 async copies (ASYNCcnt), then barrier.  src points at
// xdbl[(b*SEQ + t0)][64] ; LDS layout sBC[t - t0][0..31]  (B = 0..15, C = 16..31)
// ---------------------------------------------------------------------------
__device__ __forceinline__ void stage_bc_async(const float* __restrict__ src,
                                               float* __restrict__ sBC) {
#pragma unroll
    for (int it = 0; it < 4; ++it) {
        const int id  = it * 256 + threadIdx.x;   // 1024 x 16B segments
        const int row = id >> 3;                  // time step within chunk
        const int seg = id & 7;                   // 16B segment within 32 floats
        const float* g = src + (size_t)row * C96 + seg * 4;
        const unsigned l = (unsigned)(uintptr_t)(&sBC[row * 32 + seg * 4]);
        asm volatile("global_load_async_to_lds_b128 %0, %1, off"
                     :: "v"(l), "v"(g) : "memory");
    }
    asm volatile("s_wait_asynccnt 0x0" ::: "memory");
    __syncthreads();
}

// ---------------------------------------------------------------------------
// K3: scan pass 1 — per (b, d, chunk): zero-init local recurrence.
//     Emits P = prod_t exp(delta*A) and Hloc = local final state.
// ---------------------------------------------------------------------------
__global__ __launch_bounds__(256) void k_scan1(const float* __restrict__ x,
                                               const float* __restrict__ delta,
                                               const float* __restrict__ xdbl,
                                               const float* __restrict__ A_logs,
                                               float* __restrict__ Pw,
                                               float* __restrict__ Hloc) {
    __shared__ float sBC[CHUNK * 32];
    const int d = blockIdx.x * 256 + threadIdx.x;
    const int chunk = blockIdx.y;
    const int b = blockIdx.z;
    const int t0 = chunk * CHUNK;

    stage_bc_async(xdbl + ((size_t)b * SEQ + t0) * C96 + DTRANK, sBC);

    float A[DSTATE], h[DSTATE], P[DSTATE];
    const float4* Ar = (const float4*)(A_logs + (size_t)d * DSTATE);
#pragma unroll
    for (int q = 0; q < 4; ++q) {
        float4 t = Ar[q];
        A[4 * q + 0] = -__expf(t.x); A[4 * q + 1] = -__expf(t.y);
        A[4 * q + 2] = -__expf(t.z); A[4 * q + 3] = -__expf(t.w);
    }
#pragma unroll
    for (int n = 0; n < DSTATE; ++n) { h[n] = 0.0f; P[n] = 1.0f; }

#pragma unroll 1
    for (int t = t0; t < t0 + CHUNK; ++t) {
        const size_t row = (size_t)b * SEQ + t;
        const float u  = x[row * DINNER + d];
        const float dl = delta[row * DINNER + d];
        const float du = dl * u;
        const float4* Bp = (const float4*)&sBC[(t - t0) * 32];
#pragma unroll
        for (int q = 0; q < 4; ++q) {
            float4 Bv = Bp[q];
            float dA;
            dA = __expf(dl * A[4 * q + 0]); h[4 * q + 0] = dA * h[4 * q + 0] + du * Bv.x; P[4 * q + 0] *= dA;
            dA = __expf(dl * A[4 * q + 1]); h[4 * q + 1] = dA * h[4 * q + 1] + du * Bv.y; P[4 * q + 1] *= dA;
            dA = __expf(dl * A[4 * q + 2]); h[4 * q + 2] = dA * h[4 * q + 2] + du * Bv.z; P[4 * q + 2] *= dA;
            dA = __expf(dl * A[4 * q + 3]); h[4 * q + 3] = dA * h[4 * q + 3] + du * Bv.w; P[4 * q + 3] *= dA;
        }
    }

    const size_t base = ((size_t)(b * NCHUNK + chunk) * DSTATE) * DINNER + d;
#pragma unroll
    for (int n = 0; n < DSTATE; ++n) {
        Pw[base + (size_t)n * DINNER]   = P[n];
        Hloc[base + (size_t)n * DINNER] = h[n];
    }
}

// ---------------------------------------------------------------------------
// K4: serial combine over chunks: Hin[k] = state entering chunk k.
// ---------------------------------------------------------------------------
__global__ __launch_bounds__(256) void k_combine(const float* __restrict__ Pw,
                                                 const float* __restrict__ Hloc,
                                                 float* __restrict__ Hin) {
    const int idx = blockIdx.x * 256 + threadIdx.x;    // 0..4095
    const int b = idx / DINNER;
    const int d = idx % DINNER;
    float h[DSTATE];
#pragma unroll
    for (int n = 0; n < DSTATE; ++n) h[n] = 0.0f;
#pragma unroll 1
    for (int k = 0; k < NCHUNK; ++k) {
        const size_t base = ((size_t)(b * NCHUNK + k) * DSTATE) * DINNER + d;
#pragma unroll
        for (int n = 0; n < DSTATE; ++n) {
            const size_t o = base + (size_t)n * DINNER;
            Hin[o] = h[n];
            h[n] = Pw[o] * h[n] + Hloc[o];
        }
    }
}

// ---------------------------------------------------------------------------
// K5: scan pass 2 — start from true Hin, emit y = C.h + Ds*u.
// ---------------------------------------------------------------------------
__global__ __launch_bounds__(256) void k_scan2(const float* __restrict__ x,
                                               const float* __restrict__ delta,
                                               const float* __restrict__ xdbl,
                                               const float* __restrict__ A_logs,
                                               const float* __restrict__ Hin,
                                               const float* __restrict__ Ds,
                                               float* __restrict__ y) {
    __shared__ float sBC[CHUNK * 32];
    const int d = blockIdx.x * 256 + threadIdx.x;
    const int chunk = blockIdx.y;
    const int b = blockIdx.z;
    const int t0 = chunk * CHUNK;

    stage_bc_async(xdbl + ((size_t)b * SEQ + t0) * C96 + DTRANK, sBC);

    float A[DSTATE], h[DSTATE];
    const float4* Ar = (const float4*)(A_logs + (size_t)d * DSTATE);
#pragma unroll
    for (int q = 0; q < 4; ++q) {
        float4 t = Ar[q];
        A[4 * q + 0] = -__expf(t.x); A[4 * q + 1] = -__expf(t.y);
        A[4 * q + 2] = -__expf(t.z); A[4 * q + 3] = -__expf(t.w);
    }
    const size_t base = ((size_t)(b * NCHUNK + chunk) * DSTATE) * DINNER + d;
#pragma unroll
    for (int n = 0; n < DSTATE; ++n) h[n] = Hin[base + (size_t)n * DINNER];

    const float Dd = Ds[d];
#pragma unroll 1
    for (int t = t0; t < t0 + CHUNK; ++t) {
        const size_t row = (size_t)b * SEQ + t;
        const float u  = x[row * DINNER + d];
        const float dl = delta[row * DINNER + d];
        const float du = dl * u;
        const float4* Bp = (const float4*)&sBC[(t - t0) * 32];
        const float4* Cp = Bp + 4;
        float yv = 0.0f;
#pragma unroll
        for (int q = 0; q < 4; ++q) {
            float4 Bv = Bp[q];
            float4 Cv = Cp[q];
            float dA;
            dA = __expf(dl * A[4 * q + 0]); h[4 * q + 0] = dA * h[4 * q + 0] + du * Bv.x; yv += h[4 * q + 0] * Cv.x;
            dA = __expf(dl * A[4 * q + 1]); h[4 * q + 1] = dA * h[4 * q + 1] + du * Bv.y; yv += h[4 * q + 1] * Cv.y;
            dA = __expf(dl * A[4 * q + 2]); h[4 * q + 2] = dA * h[4 * q + 2] + du * Bv.z; yv += h[4 * q + 2] * Cv.z;
            dA = __expf(dl * A[4 * q + 3]); h[4 * q + 3] = dA * h[4 * q + 3] + du * Bv.w; yv += h[4 * q + 3] * Cv.w;
        }
        y[row * DINNER + d] = yv + Dd * u;
    }
}

// ---------------------------------------------------------------------------
extern "C" void kernel_launch(void* const* d_in, const int* in_sizes, int n_in,
                              void* d_out, int out_size, void* d_ws, size_t ws_size,
                              hipStream_t stream) {
    (void)in_sizes; (void)n_in; (void)out_size; (void)ws_size;
    const float* x      = (const float*)d_in[0];   // (2,4096,2048)
    const float* prompt = (const float*)d_in[1];   // (2,4096,16)
    const float* Wxp    = (const float*)d_in[2];   // (1,96,2048)
    const float* Wdt    = (const float*)d_in[3];   // (1,2048,64)
    const float* dtb    = (const float*)d_in[4];   // (1,2048)
    const float* A_logs = (const float*)d_in[5];   // (2048,16)
    const float* Ds     = (const float*)d_in[6];   // (2048,)
    float* y = (float*)d_out;

    float* ws    = (float*)d_ws;
    float* xdbl  = ws;                                          // 8192*96
    float* delta = xdbl + (size_t)ROWS * C96;                   // 8192*2048
    const size_t statesz = (size_t)BATCHN * NCHUNK * DSTATE * DINNER;
    float* Pw   = delta + (size_t)ROWS * DINNER;
    float* Hloc = Pw + statesz;
    float* Hin  = Hloc + statesz;

    // K1: 512 m-tiles / 8 waves per block
    k_xproj<<<ROWS / 16 / 8, 256, 0, stream>>>(x, Wxp, prompt, xdbl);
    // K2: 512 m-tiles * 16 n-groups / 8 waves per block
    k_delta<<<(ROWS / 16) * (DINNER / 128) / 8, 256, 0, stream>>>(xdbl, Wdt, dtb, delta);
    // K3/K5: grid (d-groups, chunks, batch)
    dim3 gscan(DINNER / 256, NCHUNK, BATCHN);
    k_scan1<<<gscan, 256, 0, stream>>>(x, delta, xdbl, A_logs, Pw, Hloc);
    k_combine<<<(BATCHN * DINNER) / 256, 256, 0, stream>>>(Pw, Hloc, Hin);
    k_scan2<<<gscan, 256, 0, stream>>>(x, delta, xdbl, A_logs, Hin, Ds, y);
}